// TropicalHashGrid_31001073942547
// MI455X (gfx1250) — compile-verified
//
#include <hip/hip_runtime.h>

// ---------------------------------------------------------------------------
// Instant-NGP hash-grid encode for MI455X (gfx1250, wave32).
// L=16 levels, F=2, T=19 (2^19-entry tables), N=2^21 points.
// Levels 0..4 dense (res 16,23,31,43,59), levels 5..15 hashed.
// Level-0 table (32KB) is staged to LDS via CDNA5 async global->LDS copies.
// Output stream uses non-temporal stores so the 64MB of tables stay in L2.
// ---------------------------------------------------------------------------

typedef float v2f __attribute__((ext_vector_type(2)));
typedef float v4f __attribute__((ext_vector_type(4)));

// Async global->LDS builtin expects pointers to int32x4 (per hipcc diagnostic):
typedef int g4i __attribute__((vector_size(16)));
typedef __attribute__((address_space(1))) g4i* g4i_gp;  // global side
typedef __attribute__((address_space(3))) g4i* g4i_lp;  // LDS side

#define HG_HASHMAP (1u << 19)
#define HG_MASK (HG_HASHMAP - 1u)
#define HG_P1 2654435761u
#define HG_P2 805459861u

__device__ __forceinline__ int hg_imin(int a, int b) { return a < b ? a : b; }

// ---- one hashed level: 8 gathers from global (L2-resident) ----------------
__device__ __forceinline__ void level_hash(const v2f* __restrict__ t,
                                           float xx, float yy, float zz,
                                           float s, float& a0, float& a1) {
  float px = fmaf(xx, s, 0.5f), py = fmaf(yy, s, 0.5f), pz = fmaf(zz, s, 0.5f);
  float fx = floorf(px), fy = floorf(py), fz = floorf(pz);
  float tx = px - fx, ty = py - fy, tz = pz - fz;
  unsigned ux0 = (unsigned)(int)fx, uy0 = (unsigned)(int)fy, uz0 = (unsigned)(int)fz;
  unsigned ux1 = ux0 + 1u;
  unsigned hy0 = uy0 * HG_P1, hy1 = (uy0 + 1u) * HG_P1;
  unsigned hz0 = uz0 * HG_P2, hz1 = (uz0 + 1u) * HG_P2;
  unsigned i000 = (ux0 ^ hy0 ^ hz0) & HG_MASK;
  unsigned i001 = (ux0 ^ hy0 ^ hz1) & HG_MASK;
  unsigned i010 = (ux0 ^ hy1 ^ hz0) & HG_MASK;
  unsigned i011 = (ux0 ^ hy1 ^ hz1) & HG_MASK;
  unsigned i100 = (ux1 ^ hy0 ^ hz0) & HG_MASK;
  unsigned i101 = (ux1 ^ hy0 ^ hz1) & HG_MASK;
  unsigned i110 = (ux1 ^ hy1 ^ hz0) & HG_MASK;
  unsigned i111 = (ux1 ^ hy1 ^ hz1) & HG_MASK;
  v2f f000 = t[i000], f001 = t[i001], f010 = t[i010], f011 = t[i011];
  v2f f100 = t[i100], f101 = t[i101], f110 = t[i110], f111 = t[i111];
  float wx0 = 1.f - tx, wy0 = 1.f - ty, wz0 = 1.f - tz;
  float w00 = wy0 * wz0, w01 = wy0 * tz, w10 = ty * wz0, w11 = ty * tz;
  float c000 = wx0 * w00, c001 = wx0 * w01, c010 = wx0 * w10, c011 = wx0 * w11;
  float c100 = tx * w00, c101 = tx * w01, c110 = tx * w10, c111 = tx * w11;
  a0 = fmaf(c000, f000.x, a0); a1 = fmaf(c000, f000.y, a1);
  a0 = fmaf(c001, f001.x, a0); a1 = fmaf(c001, f001.y, a1);
  a0 = fmaf(c010, f010.x, a0); a1 = fmaf(c010, f010.y, a1);
  a0 = fmaf(c011, f011.x, a0); a1 = fmaf(c011, f011.y, a1);
  a0 = fmaf(c100, f100.x, a0); a1 = fmaf(c100, f100.y, a1);
  a0 = fmaf(c101, f101.x, a0); a1 = fmaf(c101, f101.y, a1);
  a0 = fmaf(c110, f110.x, a0); a1 = fmaf(c110, f110.y, a1);
  a0 = fmaf(c111, f111.x, a0); a1 = fmaf(c111, f111.y, a1);
}

// ---- one dense level: 8 gathers, clamped stride indexing ------------------
__device__ __forceinline__ void level_dense(const v2f* __restrict__ t,
                                            float xx, float yy, float zz,
                                            float s, int res,
                                            float& a0, float& a1) {
  float px = fmaf(xx, s, 0.5f), py = fmaf(yy, s, 0.5f), pz = fmaf(zz, s, 0.5f);
  float fx = floorf(px), fy = floorf(py), fz = floorf(pz);
  float tx = px - fx, ty = py - fy, tz = pz - fz;
  int ix = (int)fx, iy = (int)fy, iz = (int)fz;
  int r1 = res - 1, rr = res * res;
  int x0 = hg_imin(ix, r1), x1 = hg_imin(ix + 1, r1);
  int y0 = hg_imin(iy, r1) * res, y1 = hg_imin(iy + 1, r1) * res;
  int z0 = hg_imin(iz, r1) * rr, z1 = hg_imin(iz + 1, r1) * rr;
  v2f f000 = t[x0 + y0 + z0], f001 = t[x0 + y0 + z1];
  v2f f010 = t[x0 + y1 + z0], f011 = t[x0 + y1 + z1];
  v2f f100 = t[x1 + y0 + z0], f101 = t[x1 + y0 + z1];
  v2f f110 = t[x1 + y1 + z0], f111 = t[x1 + y1 + z1];
  float wx0 = 1.f - tx, wy0 = 1.f - ty, wz0 = 1.f - tz;
  float w00 = wy0 * wz0, w01 = wy0 * tz, w10 = ty * wz0, w11 = ty * tz;
  float c000 = wx0 * w00, c001 = wx0 * w01, c010 = wx0 * w10, c011 = wx0 * w11;
  float c100 = tx * w00, c101 = tx * w01, c110 = tx * w10, c111 = tx * w11;
  a0 = fmaf(c000, f000.x, a0); a1 = fmaf(c000, f000.y, a1);
  a0 = fmaf(c001, f001.x, a0); a1 = fmaf(c001, f001.y, a1);
  a0 = fmaf(c010, f010.x, a0); a1 = fmaf(c010, f010.y, a1);
  a0 = fmaf(c011, f011.x, a0); a1 = fmaf(c011, f011.y, a1);
  a0 = fmaf(c100, f100.x, a0); a1 = fmaf(c100, f100.y, a1);
  a0 = fmaf(c101, f101.x, a0); a1 = fmaf(c101, f101.y, a1);
  a0 = fmaf(c110, f110.x, a0); a1 = fmaf(c110, f110.y, a1);
  a0 = fmaf(c111, f111.x, a0); a1 = fmaf(c111, f111.y, a1);
}

__global__ __launch_bounds__(256)
void TropicalHashGrid_31001073942547_kernel(const float* __restrict__ x,
                                            const float* __restrict__ table,
                                            float* __restrict__ out,
                                            int npts) {
  // Level-0 dense table: 16^3 = 4096 entries x 2 floats = 32 KB in LDS.
  __shared__ float ldsT[4096 * 2];
  {
    const v4f* __restrict__ g0 = (const v4f*)table;  // level 0 at offset 0
#if __has_builtin(__builtin_amdgcn_global_load_async_to_lds_b128)
#pragma unroll
    for (int i = 0; i < 8; ++i) {
      int e = (int)threadIdx.x + i * 256;  // 2048 x 16B total
      __builtin_amdgcn_global_load_async_to_lds_b128(
          (g4i_gp)(g0 + e), (g4i_lp)(ldsT + e * 4), 0, 0);
    }
#if __has_builtin(__builtin_amdgcn_s_wait_asynccnt)
    __builtin_amdgcn_s_wait_asynccnt(0);
#else
    asm volatile("s_wait_asynccnt 0" ::: "memory");
#endif
#else
    for (int e = (int)threadIdx.x; e < 2048; e += 256)
      ((v4f*)ldsT)[e] = g0[e];
#endif
    __syncthreads();
  }

  int n = (int)(blockIdx.x * 256 + threadIdx.x);
  if (n >= npts) return;

  // Coordinates: streamed once -> non-temporal loads.
  size_t b = (size_t)n * 3;
  float xx = __builtin_nontemporal_load(x + b + 0);
  float yy = __builtin_nontemporal_load(x + b + 1);
  float zz = __builtin_nontemporal_load(x + b + 2);

  float acc[32];
#pragma unroll
  for (int i = 0; i < 32; ++i) acc[i] = 0.f;

  const v2f* t = (const v2f*)table;
  const size_t HM = (size_t)HG_HASHMAP;

  // Level 0 from LDS (ds_load_b64 gathers).
  level_dense((const v2f*)ldsT, xx, yy, zz, 15.0f, 16, acc[0], acc[1]);
  // Dense levels 1..4 (small tables, hot in L2/L0).
  level_dense(t + 1 * HM, xx, yy, zz, 21.1106067f, 23, acc[2], acc[3]);
  level_dense(t + 2 * HM, xx, yy, zz, 29.5549328f, 31, acc[4], acc[5]);
  level_dense(t + 3 * HM, xx, yy, zz, 41.2242532f, 43, acc[6], acc[7]);
  level_dense(t + 4 * HM, xx, yy, zz, 57.3502429f, 59, acc[8], acc[9]);
  // Hashed levels 5..15 (4MB tables each; all 64MB resident in 192MB L2).
  level_hash(t + 5 * HM, xx, yy, zz, 79.6349472f, acc[10], acc[11]);
  level_hash(t + 6 * HM, xx, yy, zz, 110.430472f, acc[12], acc[13]);
  level_hash(t + 7 * HM, xx, yy, zz, 152.987203f, acc[14], acc[15]);
  level_hash(t + 8 * HM, xx, yy, zz, 211.796988f, acc[16], acc[17]);
  level_hash(t + 9 * HM, xx, yy, zz, 293.066779f, acc[18], acc[19]);
  level_hash(t + 10 * HM, xx, yy, zz, 405.374669f, acc[20], acc[21]);
  level_hash(t + 11 * HM, xx, yy, zz, 560.574389f, acc[22], acc[23]);
  level_hash(t + 12 * HM, xx, yy, zz, 775.046882f, acc[24], acc[25]);
  level_hash(t + 13 * HM, xx, yy, zz, 1071.42918f, acc[26], acc[27]);
  level_hash(t + 14 * HM, xx, yy, zz, 1481.00370f, acc[28], acc[29]);
  level_hash(t + 15 * HM, xx, yy, zz, 2047.0f, acc[30], acc[31]);

  // Output: 32 floats/point, streamed once -> non-temporal b128 stores
  // (keeps the tables resident in L2 instead of the 256MB output stream).
  v4f* o = (v4f*)(out + (size_t)n * 32);
#pragma unroll
  for (int k = 0; k < 8; ++k) {
    v4f v = {acc[4 * k + 0], acc[4 * k + 1], acc[4 * k + 2], acc[4 * k + 3]};
    __builtin_nontemporal_store(v, o + k);
  }
}

extern "C" void kernel_launch(void* const* d_in, const int* in_sizes, int n_in,
                              void* d_out, int out_size, void* d_ws, size_t ws_size,
                              hipStream_t stream) {
  const float* x = (const float*)d_in[0];      // [N,3] f32
  const float* table = (const float*)d_in[1];  // [16, 2^19, 2] f32
  float* out = (float*)d_out;                  // [N, 32] f32
  int npts = in_sizes[0] / 3;
  dim3 grid((unsigned)((npts + 255) / 256));
  TropicalHashGrid_31001073942547_kernel<<<grid, 256, 0, stream>>>(x, table, out, npts);
}